// GCN_60129542665
// MI455X (gfx1250) — compile-verified
//
#include <hip/hip_runtime.h>
#include <cstdint>

// ---------------------------------------------------------------------------
// GAT forward for MI455X (gfx1250, wave32).
//  * Node transforms: fp32 tensor path (V_WMMA_F32_16X16X4_F32), B-panel +
//    bias staged into LDS via CDNA5 async copy (global_load_async_to_lds_b32,
//    s_wait_asynccnt) so each block fetches W once.
//  * Edge softmax/aggregation: bandwidth-bound; float4 (b128) gathers and
//    global float atomics.
// ---------------------------------------------------------------------------

typedef __attribute__((ext_vector_type(2))) float v2f;
typedef __attribute__((ext_vector_type(8))) float v8f;

static constexpr int NN = 100000;   // nodes (divisible by 16)
static constexpr int EE = 1600000;  // edges (before self loops)
static constexpr int GG = 512;      // graphs
static constexpr int HH = 2;        // attention heads (all layers)

#define DEVI __device__ __forceinline__

// Monotone uint key for float atomicMax (handles negatives / -inf).
DEVI unsigned fkey(float f) {
    unsigned u = __float_as_uint(f);
    return (u & 0x80000000u) ? ~u : (u | 0x80000000u);
}
DEVI float fkeyinv(unsigned k) {
    return __uint_as_float((k & 0x80000000u) ? (k & 0x7fffffffu) : ~k);
}
static constexpr unsigned NEGINF_KEY = 0x007FFFFFu;  // fkey(-inf)

// Async copy of one 32-bit element from global memory into LDS (per-lane,
// honors EXEC). Completion tracked by ASYNCcnt.
DEVI void async_g2lds_b32(const float* gsrc, const float* ldst) {
    unsigned lds = (unsigned)(unsigned long long)ldst;          // low 32 bits = LDS byte addr
    unsigned long long g = (unsigned long long)gsrc;
    asm volatile("global_load_async_to_lds_b32 %0, %1, off" :: "v"(lds), "v"(g) : "memory");
}
DEVI void wait_asynccnt0() {
    asm volatile("s_wait_asynccnt 0x0" ::: "memory");
}

// ---------------------------------------------------------------- fills ----
__global__ void k_fill_f32(float* __restrict__ p, float v, long long n) {
    long long i = (long long)blockIdx.x * blockDim.x + threadIdx.x;
    if (i < n) p[i] = v;
}
__global__ void k_fill_u32(unsigned* __restrict__ p, unsigned v, long long n) {
    long long i = (long long)blockIdx.x * blockDim.x + threadIdx.x;
    if (i < n) p[i] = v;
}

// --------------------------------------------- degree + self-loop edge fv --
__global__ void k_deg_selfea(const int* __restrict__ dst, const float* __restrict__ ea,
                             float* __restrict__ deg, float* __restrict__ easum) {
    long long e = (long long)blockIdx.x * blockDim.x + threadIdx.x;
    if (e >= EE) return;
    int d = dst[e];
    atomicAdd(&deg[d], 1.0f);
    atomicAdd(&easum[2 * (long long)d + 0], ea[2 * e + 0]);
    atomicAdd(&easum[2 * (long long)d + 1], ea[2 * e + 1]);
}
__global__ void k_selfea_div(float* __restrict__ easum, const float* __restrict__ deg) {
    long long n = (long long)blockIdx.x * blockDim.x + threadIdx.x;
    if (n >= NN) return;
    float dg = fmaxf(deg[n], 1.0f);
    easum[2 * n + 0] /= dg;
    easum[2 * n + 1] /= dg;
}

// ------------------------------------------------- WMMA f32 GEMM + bias ----
// Y[NN x M] = X[NN x K] @ W[K x M] + bias (row-major).
// Grid: x = strip of 8 row-tiles (one per wave), y = 16-wide column tile.
// The block's B panel (Kpad x 16, zero-padded) and bias slice are staged into
// LDS once via CDNA5 async copy; each wave then runs the 16x16x4 WMMA K-loop
// reading B from LDS (bank-conflict free: lane halves hit disjoint banks).
// Lane layouts (ISA 7.12.2):
//   A (16x4): a[j] = A[lane%16][2*(lane/16)+j]
//   B (4x16): b[j] = B[2*(lane/16)+j][lane%16]
//   C/D:      acc[i] = D[i+8*(lane/16)][lane%16]
__global__ void k_wmma_gemm_bias(const float* __restrict__ X, const float* __restrict__ W,
                                 const float* __restrict__ bias, float* __restrict__ Y,
                                 int K, int M) {
    __shared__ float sB[80 * 16 + 16];          // max Kpad(80) x 16 + bias row
    const int tn = blockIdx.y;                  // column tile
    const int kBlocks = (K + 3) >> 2;
    const int panel = kBlocks * 4 * 16;         // Kpad * 16

    // 1) zero the padded panel (covers K-tail rows and M-tail columns)
    for (int i = threadIdx.x; i < panel + 16; i += blockDim.x) sB[i] = 0.f;
    __syncthreads();

    // 2) async-stage valid W elements + bias into LDS (per-lane, EXEC-masked)
    for (int i = threadIdx.x; i < panel; i += blockDim.x) {
        const int k = i >> 4;
        const int col = tn * 16 + (i & 15);
        if (k < K && col < M) async_g2lds_b32(&W[(long long)k * M + col], &sB[i]);
    }
    if (threadIdx.x < 16) {
        const int col = tn * 16 + threadIdx.x;
        if (col < M) async_g2lds_b32(&bias[col], &sB[panel + threadIdx.x]);
    }
    wait_asynccnt0();                           // this wave's async copies done
    __syncthreads();                            // all waves' copies visible

    // 3) per-wave 16x16 tile
    const int tm = blockIdx.x * (blockDim.x >> 5) + (threadIdx.x >> 5);
    if (tm >= (NN >> 4)) return;                // wave-uniform; after last barrier
    const int lane = threadIdx.x & 31;
    const int half = lane >> 4;
    const int l16 = lane & 15;
    const int row = tm * 16 + l16;

    v8f acc = {0.f, 0.f, 0.f, 0.f, 0.f, 0.f, 0.f, 0.f};
    for (int kb = 0; kb < kBlocks; ++kb) {
        const int k0 = kb * 4 + 2 * half;
        const int kS0 = (k0 < K) ? k0 : 0;      // clamped safe indices for A
        const int kS1 = (k0 + 1 < K) ? k0 + 1 : 0;
        float a0 = X[(long long)row * K + kS0];
        float a1 = X[(long long)row * K + kS1];
        if (k0 >= K)     a0 = 0.f;
        if (k0 + 1 >= K) a1 = 0.f;
        v2f a = {a0, a1};
        v2f b = {sB[k0 * 16 + l16], sB[(k0 + 1) * 16 + l16]};  // zero-padded
        acc = __builtin_amdgcn_wmma_f32_16x16x4_f32(
            /*neg_a=*/false, a, /*neg_b=*/false, b,
            /*c_mod=*/(short)0, acc, /*reuse_a=*/false, /*reuse_b=*/false);
    }
    const int col = tn * 16 + l16;
    if (col < M) {
        const float bv = sB[panel + l16];
#pragma unroll
        for (int i = 0; i < 8; ++i) {
            const int r = tm * 16 + i + 8 * half;
            Y[(long long)r * M + col] = acc[i] + bv;
        }
    }
}

// -------------------------------------------------------- edge pass 1 ------
// logit[e,h] = sum_c lrelu02(xl[s]+xr[d]+ea@We)[h,c]*att[h,c]; atomicMax lmax[d,h]
__global__ void k_edge_logit(const int* __restrict__ src, const int* __restrict__ dst,
                             const float* __restrict__ ea, const float* __restrict__ selfea,
                             const float* __restrict__ xl, const float* __restrict__ xr,
                             const float* __restrict__ We, const float* __restrict__ att,
                             float* __restrict__ logit, unsigned* __restrict__ lmaxk, int C) {
    const long long tot = (long long)(EE + NN) * HH;
    long long idx = (long long)blockIdx.x * blockDim.x + threadIdx.x;
    if (idx >= tot) return;
    const long long e = idx >> 1;
    const int h = (int)(idx & 1);
    int s, d; float e0, e1;
    if (e < EE) { s = src[e]; d = dst[e]; e0 = ea[2 * e]; e1 = ea[2 * e + 1]; }
    else        { s = d = (int)(e - EE); e0 = selfea[2 * (long long)s]; e1 = selfea[2 * (long long)s + 1]; }
    const int HC = HH * C;
    const int base = h * C;                      // multiple of 4 -> 16B aligned
    const float4* pl = (const float4*)(xl + (long long)s * HC + base);
    const float4* pr = (const float4*)(xr + (long long)d * HC + base);
    const float4* w0 = (const float4*)(We + base);
    const float4* w1 = (const float4*)(We + HC + base);
    const float4* pa = (const float4*)(att + base);
    float acc = 0.f;
    const int C4 = C >> 2;
    for (int c = 0; c < C4; ++c) {
        float4 l = pl[c], r = pr[c], u = w0[c], v = w1[c], t = pa[c];
        float m0 = l.x + r.x + e0 * u.x + e1 * v.x;
        float m1 = l.y + r.y + e0 * u.y + e1 * v.y;
        float m2 = l.z + r.z + e0 * u.z + e1 * v.z;
        float m3 = l.w + r.w + e0 * u.w + e1 * v.w;
        m0 = (m0 > 0.f) ? m0 : 0.2f * m0;
        m1 = (m1 > 0.f) ? m1 : 0.2f * m1;
        m2 = (m2 > 0.f) ? m2 : 0.2f * m2;
        m3 = (m3 > 0.f) ? m3 : 0.2f * m3;
        acc += m0 * t.x + m1 * t.y + m2 * t.z + m3 * t.w;
    }
    logit[idx] = acc;
    atomicMax(&lmaxk[(long long)d * HH + h], fkey(acc));
}

__global__ void k_lmax_decode(const unsigned* __restrict__ key, float* __restrict__ out) {
    long long i = (long long)blockIdx.x * blockDim.x + threadIdx.x;
    if (i >= (long long)NN * HH) return;
    float f = fkeyinv(key[i]);
    out[i] = (f < -3.0e38f) ? 0.0f : f;          // isneginf -> 0
}

// -------------------------------------------------------- edge pass 2 ------
__global__ void k_edge_exp(const int* __restrict__ dst, float* __restrict__ logit,
                           const float* __restrict__ lmaxf, float* __restrict__ den) {
    const long long tot = (long long)(EE + NN) * HH;
    long long idx = (long long)blockIdx.x * blockDim.x + threadIdx.x;
    if (idx >= tot) return;
    const long long e = idx >> 1;
    const int h = (int)(idx & 1);
    const int d = (e < EE) ? dst[e] : (int)(e - EE);
    float v = expf(logit[idx] - lmaxf[(long long)d * HH + h]);
    logit[idx] = v;                              // overwrite logit with ex
    atomicAdd(&den[(long long)d * HH + h], v);
}

// -------------------------------------------------------- edge pass 3 ------
// One thread per (edge, 4 channels): float4 gather of xl[s], 4 scatter-adds.
__global__ void k_edge_accum(const int* __restrict__ src, const int* __restrict__ dst,
                             const float* __restrict__ ex, const float* __restrict__ den,
                             const float* __restrict__ xl, float* __restrict__ out, int C) {
    const int HC = HH * C;
    const int Q = HC >> 2;                       // float4 groups per row
    const long long tot = (long long)(EE + NN) * Q;
    long long idx = (long long)blockIdx.x * blockDim.x + threadIdx.x;
    if (idx >= tot) return;
    const long long e = idx / Q;
    const int c = (int)(idx % Q) << 2;           // channel base (C%4==0 -> one head)
    const int h = c / C;
    int s, d;
    if (e < EE) { s = src[e]; d = dst[e]; }
    else        { s = d = (int)(e - EE); }
    const float alpha = ex[e * HH + h] / fmaxf(den[(long long)d * HH + h], 1e-16f);
    float4 v = *(const float4*)(xl + (long long)s * HC + c);
    float* o = out + (long long)d * HC + c;
    atomicAdd(o + 0, alpha * v.x);
    atomicAdd(o + 1, alpha * v.y);
    atomicAdd(o + 2, alpha * v.z);
    atomicAdd(o + 3, alpha * v.w);
}

// ------------------------------------------------------------ batch norm ---
__global__ void k_bn_stats(const float* __restrict__ h, float* __restrict__ mu,
                           float* __restrict__ inv, int HC) {
    const int c = blockIdx.x;
    __shared__ float ssum[256];
    __shared__ float ssq[256];
    float s = 0.f, q = 0.f;
    for (int n = threadIdx.x; n < NN; n += 256) {
        float v = h[(long long)n * HC + c];
        s += v; q += v * v;
    }
    ssum[threadIdx.x] = s; ssq[threadIdx.x] = q;
    __syncthreads();
    for (int off = 128; off > 0; off >>= 1) {
        if ((int)threadIdx.x < off) {
            ssum[threadIdx.x] += ssum[threadIdx.x + off];
            ssq[threadIdx.x] += ssq[threadIdx.x + off];
        }
        __syncthreads();
    }
    if (threadIdx.x == 0) {
        float m = ssum[0] / (float)NN;
        float var = ssq[0] / (float)NN - m * m;
        mu[c] = m;
        inv[c] = rsqrtf(var + 1e-5f);
    }
}

__global__ void k_bn_apply(const float* __restrict__ hin, float* __restrict__ hout,
                           const float* __restrict__ mu, const float* __restrict__ inv,
                           const float* __restrict__ g, const float* __restrict__ b, int HC) {
    const int Q = HC >> 2;
    long long idx = (long long)blockIdx.x * blockDim.x + threadIdx.x;
    if (idx >= (long long)NN * Q) return;
    const int c4 = (int)(idx % Q);
    float4 x = ((const float4*)hin)[idx];
    float4 m = ((const float4*)mu)[c4];
    float4 iv = ((const float4*)inv)[c4];
    float4 gg = ((const float4*)g)[c4];
    float4 bb = ((const float4*)b)[c4];
    float4 y;
    y.x = (x.x - m.x) * iv.x * gg.x + bb.x;
    y.y = (x.y - m.y) * iv.y * gg.y + bb.y;
    y.z = (x.z - m.z) * iv.z * gg.z + bb.z;
    y.w = (x.w - m.w) * iv.w * gg.w + bb.w;
    y.x = (y.x > 0.f) ? y.x : 0.1f * y.x;
    y.y = (y.y > 0.f) ? y.y : 0.1f * y.y;
    y.z = (y.z > 0.f) ? y.z : 0.1f * y.z;
    y.w = (y.w > 0.f) ? y.w : 0.1f * y.w;
    ((float4*)hout)[idx] = y;
}

// ---------------------------------------------------------------- pooling --
__global__ void k_pool_cnt(const int* __restrict__ batch, float* __restrict__ cnt) {
    long long n = (long long)blockIdx.x * blockDim.x + threadIdx.x;
    if (n >= NN) return;
    atomicAdd(&cnt[batch[n]], 1.0f);
}
__global__ void k_pool_sum(const float* __restrict__ h, const int* __restrict__ batch,
                           float* __restrict__ pooled) {
    long long idx = (long long)blockIdx.x * blockDim.x + threadIdx.x;  // N * 40 threads
    if (idx >= (long long)NN * 40) return;
    const long long n = idx / 40;
    const int c = (int)(idx % 40) << 2;
    float4 v = *(const float4*)(h + n * 160 + c);
    float* p = pooled + (long long)batch[n] * 160 + c;
    atomicAdd(p + 0, v.x);
    atomicAdd(p + 1, v.y);
    atomicAdd(p + 2, v.z);
    atomicAdd(p + 3, v.w);
}
__global__ void k_final(const float* __restrict__ pooled, const float* __restrict__ cnt,
                        const float* __restrict__ W, const float* __restrict__ b,
                        float* __restrict__ out) {
    long long idx = (long long)blockIdx.x * blockDim.x + threadIdx.x;
    if (idx >= (long long)GG * 2) return;
    const int g = (int)(idx >> 1);
    const int j = (int)(idx & 1);
    float s = 0.f;
    for (int c = 0; c < 160; ++c) s += pooled[(long long)g * 160 + c] * W[c * 2 + j];
    out[idx] = s / fmaxf(cnt[g], 1.0f) + b[j];
}

// ===========================================================================
extern "C" void kernel_launch(void* const* d_in, const int* in_sizes, int n_in,
                              void* d_out, int out_size, void* d_ws, size_t ws_size,
                              hipStream_t stream) {
    (void)in_sizes; (void)n_in; (void)out_size; (void)ws_size;

    const float* x     = (const float*)d_in[0];   // (N,2)
    const int*   ei    = (const int*)d_in[1];     // (2,E)
    const int*   src   = ei;
    const int*   dst   = ei + EE;
    const float* eattr = (const float*)d_in[2];   // (E,2)
    const int*   batch = (const int*)d_in[3];     // (N,)
    auto F = [&](int i) { return (const float*)d_in[i]; };
    // 4..10: g1{Wl,bl,Wr,br,We,att,bias}; 11..17: g2; 18..24: g3
    // 25,26: bn1{g,b}; 27,28: bn2; 29,30: bn3; 31,32: lin{W,b}
    // NOTE: gat 'bias' cancels exactly inside the following BatchNorm -> skipped.

    // -------- workspace layout (floats), bump-allocated --------------------
    float* ws = (float*)d_ws;
    size_t off = 0;
    auto alloc = [&](size_t n) { float* r = ws + off; off += n; return r; };
    float*    deg    = alloc(NN);
    float*    selfea = alloc(2ULL * NN);          // contiguous with deg (fill 3N)
    unsigned* lmaxk  = (unsigned*)alloc(2ULL * NN);
    float*    lmaxf  = alloc(2ULL * NN);
    float*    den    = alloc(2ULL * NN);
    float*    elog   = alloc((size_t)(EE + NN) * HH);  // logit, then ex in place
    float*    xl     = alloc((size_t)NN * 160);
    float*    xr     = alloc((size_t)NN * 160);
    float*    bufA   = alloc((size_t)NN * 160);
    float*    bufB   = alloc((size_t)NN * 160);
    float*    mu     = alloc(160);
    float*    inv    = alloc(160);
    float*    pooled = alloc((size_t)GG * 160);
    float*    cnt    = alloc(GG);                 // contiguous with pooled

    const int BT = 256;
    auto blocks = [](long long n) { return (unsigned)((n + 255) / 256); };

    // -------- self-loop edge features (mean fill) ---------------------------
    k_fill_f32<<<blocks(3LL * NN), BT, 0, stream>>>(deg, 0.f, 3LL * NN);
    k_deg_selfea<<<blocks(EE), BT, 0, stream>>>(dst, eattr, deg, selfea);
    k_selfea_div<<<blocks(NN), BT, 0, stream>>>(selfea, deg);

    // -------- one GAT layer -------------------------------------------------
    auto gat_layer = [&](const float* xin, int fin, int C,
                         const float* Wl, const float* bl,
                         const float* Wr, const float* br,
                         const float* We, const float* att, float* out) {
        const int HC = HH * C;
        const int tilesN = (HC + 15) / 16;
        const dim3 gg((unsigned)((NN / 16 + 7) / 8), (unsigned)tilesN, 1);
        k_wmma_gemm_bias<<<gg, 256, 0, stream>>>(xin, Wl, bl, xl, fin, HC);
        k_wmma_gemm_bias<<<gg, 256, 0, stream>>>(xin, Wr, br, xr, fin, HC);
        k_fill_u32<<<blocks(2LL * NN), BT, 0, stream>>>(lmaxk, NEGINF_KEY, 2LL * NN);
        k_fill_f32<<<blocks(2LL * NN), BT, 0, stream>>>(den, 0.f, 2LL * NN);
        k_fill_f32<<<blocks((long long)NN * HC), BT, 0, stream>>>(out, 0.f, (long long)NN * HC);
        const long long eh = (long long)(EE + NN) * HH;
        k_edge_logit<<<blocks(eh), BT, 0, stream>>>(src, dst, eattr, selfea, xl, xr,
                                                    We, att, elog, lmaxk, C);
        k_lmax_decode<<<blocks(2LL * NN), BT, 0, stream>>>(lmaxk, lmaxf);
        k_edge_exp<<<blocks(eh), BT, 0, stream>>>(dst, elog, lmaxf, den);
        const long long eq = (long long)(EE + NN) * (HC >> 2);
        k_edge_accum<<<blocks(eq), BT, 0, stream>>>(src, dst, elog, den, xl, out, C);
    };
    auto bn_act = [&](const float* hin, float* hout, int HC, const float* g, const float* b) {
        k_bn_stats<<<HC, 256, 0, stream>>>(hin, mu, inv, HC);
        k_bn_apply<<<blocks((long long)NN * (HC >> 2)), BT, 0, stream>>>(hin, hout, mu, inv,
                                                                         g, b, HC);
    };

    gat_layer(x,    2,  20, F(4),  F(5),  F(6),  F(7),  F(8),  F(9),  bufB);
    bn_act(bufB, bufA, 40, F(25), F(26));
    gat_layer(bufA, 40, 40, F(11), F(12), F(13), F(14), F(15), F(16), bufB);
    bn_act(bufB, bufA, 80, F(27), F(28));
    gat_layer(bufA, 80, 80, F(18), F(19), F(20), F(21), F(22), F(23), bufB);
    bn_act(bufB, bufA, 160, F(29), F(30));

    // -------- mean pool + linear head ---------------------------------------
    k_fill_f32<<<blocks((long long)GG * 160 + GG), BT, 0, stream>>>(pooled, 0.f,
                                                                    (long long)GG * 160 + GG);
    k_pool_cnt<<<blocks(NN), BT, 0, stream>>>(batch, cnt);
    k_pool_sum<<<blocks((long long)NN * 40), BT, 0, stream>>>(bufA, batch, pooled);
    k_final<<<blocks((long long)GG * 2), BT, 0, stream>>>(pooled, cnt, F(31), F(32),
                                                          (float*)d_out);
}